// Model_19636590478073
// MI455X (gfx1250) — compile-verified
//
#include <hip/hip_runtime.h>

#define DIMS     10000
#define PR_ALPHA 0.85f
#define MAX_ITER 100

typedef __attribute__((ext_vector_type(2))) float v2f;
typedef __attribute__((ext_vector_type(8))) float v8f;

// ---------------------------------------------------------------------------
// Kernel 1: PageRank power iteration, single persistent workgroup.
// LDS layout (dynamic): v[N] | acc[N] | vals[E]   (~49 KB for N=2000,E=8192)
// Faithful lax.while_loop semantics via shared 'done' flag.
// ---------------------------------------------------------------------------
__global__ __launch_bounds__(1024) void pagerank_kernel(
    const int* __restrict__ ei,   // [2,E] row-major: row=ei[e], col=ei[E+e]
    float* __restrict__ pr, int N, int E) {
  extern __shared__ float sm[];
  float* v    = sm;          // N
  float* acc  = sm + N;      // N  (also used as deg scratch)
  float* vals = sm + 2 * N;  // E
  __shared__ float errS;
  __shared__ int doneS;
  const int tid = threadIdx.x, BS = blockDim.x;
  const float p = 1.0f / (float)N;

  for (int i = tid; i < N; i += BS) acc[i] = 0.0f;
  if (tid == 0) doneS = 0;
  __syncthreads();
  // deg = bincount(col)
  for (int e = tid; e < E; e += BS) atomicAdd(&acc[ei[E + e]], 1.0f);
  __syncthreads();
  for (int e = tid; e < E; e += BS) vals[e] = PR_ALPHA / acc[ei[E + e]];
  for (int i = tid; i < N; i += BS) v[i] = p;
  __syncthreads();

  const float teleport = p * (1.0f - PR_ALPHA);
  const float thresh   = (float)N * 1e-6f;
  for (int it = 0; it < MAX_ITER; ++it) {
    for (int i = tid; i < N; i += BS) acc[i] = 0.0f;
    if (tid == 0) errS = 0.0f;
    __syncthreads();
    if (!doneS) {
      for (int e = tid; e < E; e += BS)
        atomicAdd(&acc[ei[e]], vals[e] * v[ei[E + e]]);  // segment_sum over rows
    }
    __syncthreads();
    if (!doneS) {
      float le = 0.0f;
      for (int i = tid; i < N; i += BS) {
        float vn = acc[i] + teleport;
        le += fabsf(vn - v[i]);
        v[i] = vn;
      }
      atomicAdd(&errS, le);
    }
    __syncthreads();
    if (tid == 0 && errS < thresh) doneS = 1;
    __syncthreads();
  }
  for (int i = tid; i < N; i += BS) pr[i] = v[i];
}

// ---------------------------------------------------------------------------
// Kernel 2: stable ascending rank (== position in jnp.argsort(pr)).
// node_id_hvs[i] = node_ids_weight[rank[i]]
// ---------------------------------------------------------------------------
__global__ void rank_kernel(const float* __restrict__ pr,
                            int* __restrict__ rnk, int N) {
  int i = blockIdx.x * blockDim.x + threadIdx.x;
  if (i >= N) return;
  float xi = pr[i];
  int cnt = 0;
  for (int j = 0; j < N; ++j) {
    float xj = pr[j];
    cnt += (xj < xi) || (xj == xi && j < i);
  }
  rnk[i] = cnt;
}

// ---------------------------------------------------------------------------
// Kernel 3: canonicalize edges (min,max), drop duplicate keys, pack gather
// indices rank[a] | rank[b]<<16 (sentinel 0xFFFFFFFF for duplicates).
// ---------------------------------------------------------------------------
__global__ void edge_kernel(const int* __restrict__ ei,
                            const int* __restrict__ rnk,
                            unsigned* __restrict__ packed, int N, int E) {
  int e = blockIdx.x * blockDim.x + threadIdx.x;
  if (e >= E) return;
  int r = ei[e], c = ei[E + e];
  int a = min(r, c), b = max(r, c);
  long long key = (long long)a * N + b;
  bool dup = false;
  for (int e2 = 0; e2 < e; ++e2) {
    int r2 = ei[e2], c2 = ei[E + e2];
    int a2 = min(r2, c2), b2 = max(r2, c2);
    if ((long long)a2 * N + b2 == key) { dup = true; break; }
  }
  packed[e] = dup ? 0xFFFFFFFFu
                  : ((unsigned)rnk[a] | ((unsigned)rnk[b] << 16));
}

// ---------------------------------------------------------------------------
// Kernel 4: enc[d] = sum_e A[ia_e,d]*A[ib_e,d]. Edge list staged in LDS,
// column slices of the 80 MB (L2-resident) hypervector matrix streamed.
// ---------------------------------------------------------------------------
#define EDGE_TILE 8192
__global__ __launch_bounds__(256) void enc_kernel(
    const float* __restrict__ hv, const unsigned* __restrict__ packed,
    float* __restrict__ enc, int E, int D) {
  __shared__ unsigned spk[EDGE_TILE];
  const int tid = threadIdx.x;
  const int d = blockIdx.x * blockDim.x + tid;
  float acc = 0.0f;
  for (int base = 0; base < E; base += EDGE_TILE) {
    int n = min(EDGE_TILE, E - base);
    __syncthreads();
    for (int e = tid; e < n; e += blockDim.x) spk[e] = packed[base + e];
    __syncthreads();
    if (d < D) {
      for (int e = 0; e < n; ++e) {
        unsigned pk = spk[e];
        if (pk != 0xFFFFFFFFu) {
          int ia = (int)(pk & 0xFFFFu), ib = (int)(pk >> 16);
          acc += hv[(size_t)ia * D + d] * hv[(size_t)ib * D + d];
        }
      }
    }
  }
  if (d < D) enc[d] = acc;
}

// ---------------------------------------------------------------------------
// Kernel 5: out = enc @ W^T via V_WMMA_F32_16X16X4_F32.
// A(16x4): lane L (m=L&15, khalf=(L>>4)*2) holds W[m, k+khalf .. k+khalf+1].
// B(4x16): enc chunk broadcast to all 16 N-columns (same layout transposed),
// so every column of D equals the GEMV result; column N=0 is extracted
// (lane 0 -> M=0..7 in VGPR0..7, lane 16 -> M=8..15). 8 waves split K,
// partials reduced through LDS.
// ---------------------------------------------------------------------------
__global__ __launch_bounds__(256) void gemv_wmma_kernel(
    const float* __restrict__ enc, const float* __restrict__ W,
    float* __restrict__ out, int C, int D, int out_size) {
  __shared__ float partial[8 * 16];
  const int tid   = threadIdx.x;
  const int lane  = tid & 31;
  const int wave  = tid >> 5;
  const int m     = lane & 15;
  const int khalf = (lane >> 4) * 2;

  v8f cacc = {};
  for (int k = wave * 4; k + 3 < D; k += 8 * 4) {
    v2f a, b;
    if (m < C) {
      a.x = W[(size_t)m * D + k + khalf];
      a.y = W[(size_t)m * D + k + khalf + 1];
    } else {
      a.x = 0.0f; a.y = 0.0f;
    }
    b.x = enc[k + khalf];
    b.y = enc[k + khalf + 1];
    cacc = __builtin_amdgcn_wmma_f32_16x16x4_f32(
        /*neg_a=*/false, a, /*neg_b=*/false, b,
        /*c_mod=*/(short)0, cacc, /*reuse_a=*/false, /*reuse_b=*/false);
  }

  if (lane == 0) {
#pragma unroll
    for (int r = 0; r < 8; ++r) partial[wave * 16 + r] = cacc[r];       // M=0..7
  } else if (lane == 16) {
#pragma unroll
    for (int r = 0; r < 8; ++r) partial[wave * 16 + 8 + r] = cacc[r];   // M=8..15
  }
  __syncthreads();
  if (tid < 16) {
    float s = 0.0f;
#pragma unroll
    for (int w = 0; w < 8; ++w) s += partial[w * 16 + tid];
    if (tid < C && tid < out_size) out[tid] = s;
  }
  for (int i = C + tid; i < out_size; i += blockDim.x) out[i] = 0.0f;
}

// ---------------------------------------------------------------------------
extern "C" void kernel_launch(void* const* d_in, const int* in_sizes, int n_in,
                              void* d_out, int out_size, void* d_ws, size_t ws_size,
                              hipStream_t stream) {
  const int*   ei = (const int*)d_in[0];     // edge_index [2,E] int32
  const float* hv = (const float*)d_in[1];   // node_ids_weight [N,D] f32
  const float* W  = (const float*)d_in[2];   // classify_weight [C,D] f32
  const int E = in_sizes[0] / 2;
  const int N = in_sizes[1] / DIMS;
  const int C = in_sizes[2] / DIMS;

  char* ws = (char*)d_ws;
  float*    pr     = (float*)ws;
  int*      rnk    = (int*)(ws + sizeof(float) * N);
  unsigned* packed = (unsigned*)(ws + sizeof(float) * N + sizeof(int) * N);
  float*    enc    = (float*)(ws + sizeof(float) * N + sizeof(int) * N +
                              sizeof(unsigned) * E);

  size_t lds_bytes = (size_t)(2 * N + E) * sizeof(float);
  pagerank_kernel<<<1, 1024, lds_bytes, stream>>>(ei, pr, N, E);
  rank_kernel<<<(N + 255) / 256, 256, 0, stream>>>(pr, rnk, N);
  edge_kernel<<<(E + 255) / 256, 256, 0, stream>>>(ei, rnk, packed, N, E);
  enc_kernel<<<(DIMS + 255) / 256, 256, 0, stream>>>(hv, packed, enc, E, DIMS);
  gemv_wmma_kernel<<<1, 256, 0, stream>>>(enc, W, (float*)d_out, C, DIMS, out_size);
}